// Sabre_48524540510378
// MI455X (gfx1250) — compile-verified
//
#include <hip/hip_runtime.h>

typedef float v2f __attribute__((ext_vector_type(2)));
typedef float v8f __attribute__((ext_vector_type(8)));

#define EPS_C    0.03137254901960784f   /* 8/255 */
#define ANNEAL_C 1e-12f
#define ERRC_C   0.001f
#define NPX      50176                  /* 224*224 */
#define NCH      96                     /* 32*3 channel-images */

/* KD[f][t] = DEC_f[5-t] : correlation-orientation taps; also equals REC_f[t]. */
__device__ const float KD[2][6] = {
  {-0.0727326195128539f, 0.3378976624578092f, 0.8525720202122554f,
    0.38486484686420286f, -0.0727326195128539f, -0.01565572813546454f},
  {-0.01565572813546454f, 0.0727326195128539f, 0.38486484686420286f,
   -0.8525720202122554f, 0.3378976624578092f, 0.0727326195128539f}};

__device__ __forceinline__ int clamp05(int t) { return t < 0 ? 0 : (t > 5 ? 5 : t); }

/* branchless symmetric reflection into [0,N): two chained selects (no exec divergence).
 * if i<0 then -1-i is in [0,4] < N, so the second select composes correctly. */
__device__ __forceinline__ int symref(int i, int N) {
  i = (i < 0) ? (-1 - i) : i;
  i = (i >= N) ? (2 * N - 1 - i) : i;
  return i;
}

/* ---------------- variant selection ---------------- */
__global__ void k_variant_stats(const float* __restrict__ x,
                                const float* __restrict__ nz,
                                int* __restrict__ best) {
  int bc = blockIdx.x, tid = threadIdx.x;
  float s[10], q[10];
#pragma unroll
  for (int k = 0; k < 10; k++) { s[k] = 0.f; q[k] = 0.f; }
  const float* xp = x + (size_t)bc * NPX;
  const float* np = nz + (size_t)bc * 10 * NPX;
  for (int px = tid; px < NPX; px += 256) {
    float xv = xp[px];
    float xt = 0.1f * xv + 0.9f * (rintf(xv * 10.f) * 0.1f);
#pragma unroll
    for (int k = 0; k < 10; k++) {
      float u = np[k * NPX + px];
      float v = xt + (2.f * u - 1.f) * EPS_C * ((float)k * (1.f / 9.f));
      v = fminf(fmaxf(v, 0.f), 1.f);
      s[k] += v; q[k] += v * v;
    }
  }
  __shared__ float red[256];
  __shared__ float fs[10], fq[10];
  for (int k = 0; k < 10; k++) {
    red[tid] = s[k]; __syncthreads();
    for (int o = 128; o > 0; o >>= 1) { if (tid < o) red[tid] += red[tid + o]; __syncthreads(); }
    if (tid == 0) fs[k] = red[0];
    __syncthreads();
    red[tid] = q[k]; __syncthreads();
    for (int o = 128; o > 0; o >>= 1) { if (tid < o) red[tid] += red[tid + o]; __syncthreads(); }
    if (tid == 0) fq[k] = red[0];
    __syncthreads();
  }
  if (tid == 0) {
    int bi = 0; float bv = 3.4e38f;
    for (int k = 0; k < 10; k++) {
      float var = fq[k] - fs[k] * fs[k] * (1.f / (float)NPX); /* monotone in std */
      if (var < bv) { bv = var; bi = k; }
    }
    best[bc] = bi;
  }
}

__global__ void k_select(const float* __restrict__ x, const float* __restrict__ nz,
                         const int* __restrict__ best, float* __restrict__ xsel) {
  int gid = blockIdx.x * 256 + threadIdx.x;
  if (gid >= NCH * NPX) return;
  int bc = gid / NPX, px = gid - bc * NPX;
  int k = best[bc];
  float xv = x[gid];
  float xt = 0.1f * xv + 0.9f * (rintf(xv * 10.f) * 0.1f);
  float u = nz[((size_t)bc * 10 + k) * NPX + px];
  float v = xt + (2.f * u - 1.f) * EPS_C * ((float)k * (1.f / 9.f));
  xsel[gid] = fminf(fmaxf(v, 0.f), 1.f);
}

/* ---------------- analysis filter bank (WMMA f32 16x16x4, K=20 chained) ----------------
 * D[(f,do), row] = sum_u A[(f,do),u] * B[u,row],  A[(f,do),u]=KD[f][u-2do], B[u,row]=xe(2*o0+u)
 * 16 rows x (2 filters x 8 outputs) per wave-tile. All loads address-safe; masks via select
 * so EXEC stays all-ones (WMMA requirement) and no divergent branches appear in the loop. */
__global__ __launch_bounds__(32)
void k_afb(const float* __restrict__ in, float* __restrict__ out,
           int R1, int R0, int ti2, int ti1, int ti0, int tiq,
           int N, int Wo, int so2, int so1, int so0, int sf, int soo, int oTiles) {
  int lane = threadIdx.x;
  int to = blockIdx.x % oTiles, tr = blockIdx.x / oTiles;
  int o0 = to * 8;
  int Rtot = NCH * R1 * R0;
  int rowN = tr * 16 + (lane & 15);
  bool rowOK = rowN < Rtot;
  int row = rowOK ? rowN : (Rtot - 1);
  int r0 = row % R0; int rr = row / R0; int r1 = rr % R1; int r2 = rr / R1;
  const float* rp = in + r2 * ti2 + r1 * ti1 + r0 * ti0;
  int hf = lane >> 4;
  int m = lane & 15, f = m >> 3, d = m & 7;
  v8f acc = {0.f,0.f,0.f,0.f,0.f,0.f,0.f,0.f};
#pragma unroll
  for (int cu = 0; cu < 5; cu++) {
    int ub = 4 * cu + 2 * hf;
    int t0 = ub - 2 * d, t1 = t0 + 1;
    float a0 = KD[f][clamp05(t0)];          /* always-valid load */
    float a1 = KD[f][clamp05(t1)];
    a0 = (t0 >= 0 && t0 <= 5) ? a0 : 0.f;   /* mask via cndmask  */
    a1 = (t1 >= 0 && t1 <= 5) ? a1 : 0.f;
    int p  = 2 * o0 + ub - 4;               /* symmetric extension: left pad 4 */
    int i0 = symref(p, N);
    int i1 = symref(p + 1, N);
    v2f a, b;
    a.x = a0; a.y = a1;
    b.x = rp[i0 * tiq];
    b.y = rp[i1 * tiq];
    acc = __builtin_amdgcn_wmma_f32_16x16x4_f32(false, a, false, b, (short)0, acc, false, false);
  }
  float* op = out + r2 * so2 + r1 * so1 + r0 * so0;
#pragma unroll
  for (int v = 0; v < 8; v++) {
    int mm = v + 8 * hf, ff = mm >> 3, dd = mm & 7, o = o0 + dd;
    if (rowOK && o < Wo) op[ff * sf + o * soo] = acc[v];
  }
}

/* ---------------- synthesis filter bank (WMMA f32 16x16x4, K=20 chained) ----------------
 * y[m,row] = sum_{f,q} KD[f][m+4-2q] * src_f[row,q]; 16 outputs x 16 rows per tile.
 * Null lo/hi pointers are uniform (scalar branch); per-lane q range masked via select. */
__global__ __launch_bounds__(32)
void k_sfb(const float* __restrict__ lo, const float* __restrict__ hi,
           float* __restrict__ out, int R1, int R0,
           int tl2, int tl1, int tl0, int tlq,
           int th2, int th1, int th0, int thq,
           int n, int Mout, int so2, int so1, int so0, int som, int mTiles) {
  int lane = threadIdx.x;
  int tm = blockIdx.x % mTiles, tr = blockIdx.x / mTiles;
  int m0 = tm * 16;
  int Rtot = NCH * R1 * R0;
  int rowN = tr * 16 + (lane & 15);
  bool rowOK = rowN < Rtot;
  int row = rowOK ? rowN : (Rtot - 1);
  int r0 = row % R0; int rr = row / R0; int r1 = rr % R1; int r2 = rr / R1;
  const float* lp = lo ? (lo + r2 * tl2 + r1 * tl1 + r0 * tl0) : (const float*)0;
  const float* hp = hi ? (hi + r2 * th2 + r1 * th1 + r0 * th0) : (const float*)0;
  int hf = lane >> 4;
  int mm = lane & 15;
  int q0 = m0 >> 1;
  v8f acc = {0.f,0.f,0.f,0.f,0.f,0.f,0.f,0.f};
#pragma unroll
  for (int cu = 0; cu < 5; cu++) {
    int dq = 2 * cu + hf;
    int tau = mm + 4 - 2 * dq;
    float a0 = KD[0][clamp05(tau)];
    float a1 = KD[1][clamp05(tau)];
    bool tv = (tau >= 0 && tau <= 5);
    a0 = tv ? a0 : 0.f;
    a1 = tv ? a1 : 0.f;
    int q = q0 + dq;
    int qc = q < n ? q : (n - 1);           /* always-valid index */
    float bl = 0.f, bh = 0.f;
    if (lp) bl = lp[qc * tlq];              /* uniform null check -> scalar branch */
    if (hp) bh = hp[qc * thq];
    bool qv = q < n;
    v2f a, b;
    a.x = a0; a.y = a1;
    b.x = qv ? bl : 0.f;
    b.y = qv ? bh : 0.f;
    acc = __builtin_amdgcn_wmma_f32_16x16x4_f32(false, a, false, b, (short)0, acc, false, false);
  }
  float* op = out + r2 * so2 + r1 * so1 + r0 * so0;
#pragma unroll
  for (int v = 0; v < 8; v++) {
    int mv = m0 + v + 8 * hf;
    if (rowOK && mv < Mout) op[mv * som] = acc[v];
  }
}

/* ---------------- energies -> thr ---------------- */
__global__ void k_thr(const float* __restrict__ C1, const float* __restrict__ C2,
                      const float* __restrict__ C3, const float* __restrict__ C4,
                      const float* __restrict__ lam, float* __restrict__ thr) {
  int c = blockIdx.x, tid = threadIdx.x;
  __shared__ float red[256];
  __shared__ float en[13];
  const float* bp[13]; int cn[13];
  const float* b1 = C1 + (size_t)c * 4 * 12996;
  const float* b2 = C2 + (size_t)c * 4 * 3481;
  const float* b3 = C3 + (size_t)c * 4 * 1024;
  const float* b4 = C4 + (size_t)c * 4 * 324;
  bp[0] = b4;            cn[0] = 324;
  bp[1] = b1 + 12996;    cn[1] = 12996;
  bp[2] = b1 + 2*12996;  cn[2] = 12996;
  bp[3] = b1 + 3*12996;  cn[3] = 12996;
  bp[4] = b2 + 3481;     cn[4] = 3481;
  bp[5] = b2 + 2*3481;   cn[5] = 3481;
  bp[6] = b2 + 3*3481;   cn[6] = 3481;
  bp[7] = b3 + 1024;     cn[7] = 1024;
  bp[8] = b3 + 2*1024;   cn[8] = 1024;
  bp[9] = b3 + 3*1024;   cn[9] = 1024;
  bp[10] = b4 + 324;     cn[10] = 324;
  bp[11] = b4 + 2*324;   cn[11] = 324;
  bp[12] = b4 + 3*324;   cn[12] = 324;
  for (int i = 0; i < 13; i++) {
    float ss = 0.f; const float* p = bp[i]; int nn = cn[i];
    for (int j = tid; j < nn; j += 256) { float v = p[j]; ss += v * v; }
    red[tid] = ss; __syncthreads();
    for (int o = 128; o > 0; o >>= 1) { if (tid < o) red[tid] += red[tid + o]; __syncthreads(); }
    if (tid == 0) en[i] = red[0];
    __syncthreads();
  }
  if (tid == 0) {
    float mn = en[0], mx = en[0];
    for (int i = 1; i < 13; i++) { mn = fminf(mn, en[i]); mx = fmaxf(mx, en[i]); }
    float den = mx - mn + ERRC_C;
    float mean = 0.f;
    for (int i = 0; i < 13; i++) { en[i] = (en[i] - mn) / den; mean += en[i]; }
    mean *= (1.f / 13.f);
    float var = 0.f;
    for (int i = 0; i < 13; i++) { float d = en[i] - mean; var += d * d; }
    float sd = sqrtf(var / 12.f);                     /* ddof=1 */
    float t = 0.25f * expf(-lam[0] * mean / (sd + ERRC_C));  /* 1/J, J=4 */
    thr[c] = fminf(fmaxf(t, 0.f), 1.f);
  }
}

/* ---------------- radix select: value at descending rank sel ---------------- */
__global__ void k_radix(const float* __restrict__ C1, const float* __restrict__ C2,
                        const float* __restrict__ C3, const float* __restrict__ C4,
                        const float* __restrict__ thr, float* __restrict__ bthr) {
  int band = blockIdx.x % 5, c = blockIdx.x / 5, tid = threadIdx.x;
  const float* p; int cnt;
  switch (band) {
    case 0: p = C4 + (size_t)c * 4 * 324;            cnt = 324;      break;
    case 1: p = C1 + (size_t)c * 4 * 12996 + 12996;  cnt = 3*12996;  break;
    case 2: p = C2 + (size_t)c * 4 * 3481 + 3481;    cnt = 3*3481;   break;
    case 3: p = C3 + (size_t)c * 4 * 1024 + 1024;    cnt = 3*1024;   break;
    default: p = C4 + (size_t)c * 4 * 324 + 324;     cnt = 3*324;    break;
  }
  int sel = (int)floorf(thr[c] * (float)(cnt - 1));
  __shared__ unsigned hist[256];
  __shared__ int srem;
  __shared__ unsigned spre;
  if (tid == 0) { srem = sel; spre = 0u; }
  __syncthreads();
  for (int shift = 24; shift >= 0; shift -= 8) {
    hist[tid] = 0u;
    __syncthreads();
    unsigned pre = spre;
    unsigned km = (shift == 24) ? 0u : (0xFFFFFFFFu << (shift + 8));
    for (int i = tid; i < cnt; i += 256) {
      unsigned b = __float_as_uint(fabsf(p[i]));
      if ((b & km) == pre) atomicAdd(&hist[(b >> shift) & 255u], 1u);
    }
    __syncthreads();
    if (tid == 0) {
      int rem = srem;
      for (int bin = 255; bin >= 0; bin--) {
        int hc = (int)hist[bin];
        if (rem < hc) { spre = pre | ((unsigned)bin << shift); break; }
        rem -= hc;
      }
      srem = rem;
    }
    __syncthreads();
  }
  if (tid == 0) bthr[c * 5 + band] = __uint_as_float(spre);
}

__global__ void k_apply(float* __restrict__ base, int gsz, int cstride, int soff,
                        const float* __restrict__ bthr, int band) {
  int gid = blockIdx.x * 256 + threadIdx.x;
  if (gid >= NCH * gsz) return;
  int c = gid / gsz, r = gid - c * gsz;
  float* pp = base + (size_t)c * cstride + soff + r;
  float v = *pp;
  if (fabsf(v) < bthr[c * 5 + band]) *pp = v * ANNEAL_C;
}

/* ---------------- combine (ANNEAL*F + X) + per-channel min/max, then normalize ------- */
__global__ void k_combine(float* __restrict__ out, float* __restrict__ mn, float* __restrict__ mx) {
  int ch = blockIdx.x, tid = threadIdx.x;
  int b = ch / 42, r = ch % 42, c = r / 14, j = r % 14;
  int cp = b * 3 + c;
  float* slice = out + (size_t)(cp * 14 + j) * NPX;
  const float* Fs = out + (size_t)(cp * 14 + 13) * NPX;
  float lmn = 3.4e38f, lmx = -3.4e38f;
  for (int i = tid; i < NPX; i += 256) {
    float Fv = Fs[i];
    float v;
    if (j == 13) v = Fv;                       /* slice 13 stays F (read-only) */
    else { v = slice[i] + ANNEAL_C * Fv; slice[i] = v; } /* (1-1e-12)==1 in f32 */
    lmn = fminf(lmn, v); lmx = fmaxf(lmx, v);
  }
  __shared__ float rn[256], rx[256];
  rn[tid] = lmn; rx[tid] = lmx; __syncthreads();
  for (int o = 128; o > 0; o >>= 1) {
    if (tid < o) { rn[tid] = fminf(rn[tid], rn[tid + o]); rx[tid] = fmaxf(rx[tid], rx[tid + o]); }
    __syncthreads();
  }
  if (tid == 0) { mn[ch] = rn[0]; mx[ch] = rx[0]; }
}

__global__ void k_norm(float* __restrict__ out, const float* __restrict__ mn,
                       const float* __restrict__ mx, int total) {
  int gid = blockIdx.x * 256 + threadIdx.x;
  if (gid >= total) return;
  int ch = gid / NPX;
  out[gid] = (out[gid] - mn[ch]) / (mx[ch] - mn[ch]);
}

/* ---------------- host-side helpers ---------------- */
static void run_afb(hipStream_t st, const float* in, float* out,
                    int R1, int R0, int ti2, int ti1, int ti0, int tiq,
                    int N, int Wo, int so2, int so1, int so0, int sf, int soo) {
  int Rtot = NCH * R1 * R0;
  int rT = (Rtot + 15) / 16, oT = (Wo + 7) / 8;
  k_afb<<<rT * oT, 32, 0, st>>>(in, out, R1, R0, ti2, ti1, ti0, tiq, N, Wo,
                                so2, so1, so0, sf, soo, oT);
}
static void run_sfb(hipStream_t st, const float* lo, const float* hi, float* out,
                    int R1, int R0, int tl2, int tl1, int tl0, int tlq,
                    int th2, int th1, int th0, int thq,
                    int n, int Mout, int so2, int so1, int so0, int som) {
  int Rtot = NCH * R1 * R0;
  int rT = (Rtot + 15) / 16, mT = (Mout + 15) / 16;
  k_sfb<<<rT * mT, 32, 0, st>>>(lo, hi, out, R1, R0, tl2, tl1, tl0, tlq,
                                th2, th1, th0, thq, n, Mout, so2, so1, so0, som, mT);
}

extern "C" void kernel_launch(void* const* d_in, const int* in_sizes, int n_in,
                              void* d_out, int out_size, void* d_ws, size_t ws_size,
                              hipStream_t stream) {
  const float* x   = (const float*)d_in[0];
  const float* nz  = (const float*)d_in[1];
  const float* lam = (const float*)d_in[2];
  float* out = (float*)d_out;
  float* ws  = (float*)d_ws;

  /* workspace layout (floats) */
  const size_t O_XSEL = 0;               /* 96*50176            */
  const size_t O_C1   = 4816896;         /* 96*4*12996          */
  const size_t O_C2   = 9807360;         /* 96*4*3481           */
  const size_t O_C3   = 11144064;        /* 96*4*1024           */
  const size_t O_C4   = 11537280;        /* 96*4*324            */
  const size_t O_ROW  = 11661696;        /* 96*2*114*224        */
  const size_t O_LOW  = O_ROW;           /* reuse after analysis */
  const size_t O_HIGH = O_ROW + 2451456;
  const size_t O_LLA  = 16564608;        /* 96*114*114          */
  const size_t O_LLB  = 17812224;
  const size_t O_THR  = 19059840;
  const size_t O_BTHR = 19059936;
  const size_t O_BEST = 19060416;
  const size_t O_MN   = 19060512;
  const size_t O_MX   = 19061856;

  float* xsel = ws + O_XSEL;
  float* C1 = ws + O_C1; float* C2 = ws + O_C2; float* C3 = ws + O_C3; float* C4 = ws + O_C4;
  float* rowt = ws + O_ROW;
  float* lowb = ws + O_LOW; float* highb = ws + O_HIGH;
  float* pingA = ws + O_LLA; float* pingB = ws + O_LLB;
  float* thr = ws + O_THR; float* bthr = ws + O_BTHR;
  int* best = (int*)(ws + O_BEST);
  float* mnp = ws + O_MN; float* mxp = ws + O_MX;

  /* 1) variant selection */
  k_variant_stats<<<NCH, 256, 0, stream>>>(x, nz, best);
  k_select<<<(NCH * NPX + 255) / 256, 256, 0, stream>>>(x, nz, best, xsel);

  /* 2) forward DWT, 4 levels (row pass writes transposed, col pass writes subbands) */
  run_afb(stream, xsel, rowt, 1, 224, NPX, 0, 224, 1, 224, 114, 2*114*224, 0, 1, 114*224, 224);
  run_afb(stream, rowt, C1, 2, 114, 2*114*224, 114*224, 224, 1, 224, 114, 4*12996, 12996, 1, 2*12996, 114);
  run_afb(stream, C1, rowt, 1, 114, 4*12996, 0, 114, 1, 114, 59, 2*59*114, 0, 1, 59*114, 114);
  run_afb(stream, rowt, C2, 2, 59, 2*59*114, 59*114, 114, 1, 114, 59, 4*3481, 3481, 1, 2*3481, 59);
  run_afb(stream, C2, rowt, 1, 59, 4*3481, 0, 59, 1, 59, 32, 2*32*59, 0, 1, 32*59, 59);
  run_afb(stream, rowt, C3, 2, 32, 2*32*59, 32*59, 59, 1, 59, 32, 4*1024, 1024, 1, 2*1024, 32);
  run_afb(stream, C3, rowt, 1, 32, 4*1024, 0, 32, 1, 32, 18, 2*18*32, 0, 1, 18*32, 32);
  run_afb(stream, rowt, C4, 2, 18, 2*18*32, 18*32, 32, 1, 32, 18, 4*324, 324, 1, 2*324, 18);

  /* 3) thresholds: energies -> thr -> radix-select bthr -> anneal small coeffs */
  k_thr<<<NCH, 256, 0, stream>>>(C1, C2, C3, C4, lam, thr);
  k_radix<<<NCH * 5, 256, 0, stream>>>(C1, C2, C3, C4, thr, bthr);
  k_apply<<<(NCH * 324 + 255) / 256, 256, 0, stream>>>(C4, 324, 4*324, 0, bthr, 0);
  k_apply<<<(NCH * 3*12996 + 255) / 256, 256, 0, stream>>>(C1, 3*12996, 4*12996, 12996, bthr, 1);
  k_apply<<<(NCH * 3*3481 + 255) / 256, 256, 0, stream>>>(C2, 3*3481, 4*3481, 3481, bthr, 2);
  k_apply<<<(NCH * 3*1024 + 255) / 256, 256, 0, stream>>>(C3, 3*1024, 4*1024, 1024, bthr, 3);
  k_apply<<<(NCH * 3*324 + 255) / 256, 256, 0, stream>>>(C4, 3*324, 4*324, 324, bthr, 4);

  /* 4) reconstructions: outs[j] = ANNEAL*F + (1-ANNEAL)*X_j by linearity of iwt */
  const int dims[5] = {224, 114, 59, 32, 18};
  const int SZs[5]  = {0, 12996, 3481, 1024, 324};
  float* CL[5] = {nullptr, C1, C2, C3, C4};

  auto idwt = [&](int D, const float* ll, int l2, int l0, int lq,
                  const float* LH, const float* HL, const float* HH, int cs2,
                  float* dst, int os2, int os0, int osm) {
    int H2 = 2 * D - 4;
    bool hL = (ll != nullptr) || (HL != nullptr);
    bool hH = (LH != nullptr) || (HH != nullptr);
    if (hL) run_sfb(stream, ll, HL, lowb, 1, D, l2, 0, l0, lq, cs2, 0, 1, D,
                    D, H2, H2 * D, 0, 1, D);
    if (hH) run_sfb(stream, LH, HH, highb, 1, D, cs2, 0, 1, D, cs2, 0, 1, D,
                    D, H2, H2 * D, 0, 1, D);
    run_sfb(stream, hL ? lowb : nullptr, hH ? highb : nullptr, dst, 1, H2,
            H2 * D, 0, D, 1, H2 * D, 0, D, 1, D, H2, os2, 0, os0, osm);
  };

  /* chains: j<12 single-band B_ik, j=12 approx-only A, j=13 full F */
  for (int j = 0; j < 14; j++) {
    int startLev, kk;
    bool useApprox, fullDet;
    if (j < 12) { startLev = j / 3 + 1; kk = j % 3; useApprox = false; fullDet = false; }
    else if (j == 12) { startLev = 4; kk = -1; useApprox = true; fullDet = false; }
    else { startLev = 4; kk = -1; useApprox = true; fullDet = true; }
    const float* ll = nullptr; int l2 = 0, l0 = 0, lq = 0;
    if (useApprox) { ll = C4; l2 = 4 * 324; l0 = 1; lq = 18; }
    float* pp = pingA; float* qq = pingB;
    for (int L = startLev; L >= 1; L--) {
      int D = dims[L], SZ = SZs[L], H2 = 2 * D - 4;
      const float* Cb = CL[L];
      const float* LH = nullptr; const float* HL = nullptr; const float* HH = nullptr;
      if (fullDet) { LH = Cb + SZ; HL = Cb + 2 * SZ; HH = Cb + 3 * SZ; }
      else if (L == startLev && kk >= 0) {
        if (kk == 0) LH = Cb + SZ;
        else if (kk == 1) HL = Cb + 2 * SZ;
        else HH = Cb + 3 * SZ;
      }
      float* dst; int os2, os0, osm;
      if (L > 1) { dst = pp; os2 = H2 * H2; os0 = H2; osm = 1; }
      else { dst = out + (size_t)j * NPX; os2 = 14 * NPX; os0 = 224; osm = 1; }
      idwt(D, ll, l2, l0, lq, LH, HL, HH, 4 * SZ, dst, os2, os0, osm);
      ll = pp; l2 = H2 * H2; l0 = 1; lq = H2;
      float* t = pp; pp = qq; qq = t;
    }
  }

  /* 5) combine + per-channel minmax, then normalize */
  k_combine<<<32 * 42, 256, 0, stream>>>(out, mnp, mxp);
  k_norm<<<(out_size + 255) / 256, 256, 0, stream>>>(out, mnp, mxp, out_size);
}